// GlobalModel_12077448036507
// MI455X (gfx1250) — compile-verified
//
#include <hip/hip_runtime.h>
#include <hip/hip_bf16.h>

#define DFEAT 128
#define IN_DIM 384

typedef __attribute__((ext_vector_type(2))) float v2f;
typedef __attribute__((ext_vector_type(4))) float v4f;
typedef __attribute__((ext_vector_type(8))) float v8f;

// ---------------------------------------------------------------------------
// Kernel 1: init workspace inp[G][384]: cols 0..255 = 0, cols 256..383 = u
// ---------------------------------------------------------------------------
__global__ __launch_bounds__(256) void init_inp(const float* __restrict__ u,
                                                float* __restrict__ inp, int total) {
  int i = blockIdx.x * 256 + threadIdx.x;
  if (i >= total) return;
  int g = i / IN_DIM;
  int c = i - g * IN_DIM;
  inp[i] = (c >= 2 * DFEAT) ? u[g * DFEAT + (c - 2 * DFEAT)] : 0.0f;
}

// ---------------------------------------------------------------------------
// Kernel 2: node scatter-sum. One wave per row (lane holds 4 consecutive
// floats). batch is sorted -> 8 rows per block usually share one graph id:
// LDS tree-reduce 8 rows -> 1 before atomics (8x fewer L2 atomics).
// ---------------------------------------------------------------------------
__global__ __launch_bounds__(256) void node_scatter(const float* __restrict__ x,
                                                    const int* __restrict__ batch,
                                                    float* __restrict__ inp,
                                                    int n_nodes) {
  __shared__ float s[8][DFEAT];
  __shared__ int sg[8];
  const int wave = threadIdx.x >> 5;
  const int lane = threadIdx.x & 31;
  const long long row = (long long)blockIdx.x * 8 + wave;

  v4f v = {0.f, 0.f, 0.f, 0.f};
  int g = -1;
  if (row < n_nodes) {
    const v4f* p = (const v4f*)(x + row * (long long)DFEAT);
    v = __builtin_nontemporal_load(p + lane);   // streaming: don't pollute L2
    g = batch[row];
  }
  *(v4f*)&s[wave][lane * 4] = v;
  if (lane == 0) sg[wave] = g;
  __syncthreads();

  bool uniform = (sg[0] >= 0);
#pragma unroll
  for (int i = 1; i < 8; ++i) uniform = uniform && (sg[i] == sg[0]);

  if (uniform) {
    // tree-reduce 8 rows down to row 0 in LDS
#pragma unroll
    for (int stride = 4; stride >= 1; stride >>= 1) {
      if (wave < stride) {
        v4f a = *(v4f*)&s[wave][lane * 4];
        v4f c = *(v4f*)&s[wave + stride][lane * 4];
        a += c;
        *(v4f*)&s[wave][lane * 4] = a;
      }
      __syncthreads();
    }
    if (wave == 0) {
      v4f a = *(v4f*)&s[0][lane * 4];
      float* dst = inp + (size_t)sg[0] * IN_DIM + lane * 4;
      atomicAdd(dst + 0, a.x);
      atomicAdd(dst + 1, a.y);
      atomicAdd(dst + 2, a.z);
      atomicAdd(dst + 3, a.w);
    }
  } else if (g >= 0) {
    float* dst = inp + (size_t)g * IN_DIM + lane * 4;
    atomicAdd(dst + 0, v.x);
    atomicAdd(dst + 1, v.y);
    atomicAdd(dst + 2, v.z);
    atomicAdd(dst + 3, v.w);
  }
}

// ---------------------------------------------------------------------------
// Kernel 3: edge scatter-sum. Graph ids are random -> direct L2 atomics into
// the 1.5MB L2-resident accumulator (cols 128..255).
// ---------------------------------------------------------------------------
__global__ __launch_bounds__(256) void edge_scatter(const float* __restrict__ ea,
                                                    const int* __restrict__ edge_src,
                                                    const int* __restrict__ batch,
                                                    float* __restrict__ inp,
                                                    int n_edges) {
  const int wave = threadIdx.x >> 5;
  const int lane = threadIdx.x & 31;
  const long long e = (long long)blockIdx.x * 8 + wave;
  if (e >= n_edges) return;
  const int src = edge_src[e];        // edge_index row 0
  const int g = batch[src];
  const v4f* p = (const v4f*)(ea + e * (long long)DFEAT);
  v4f v = __builtin_nontemporal_load(p + lane);
  float* dst = inp + (size_t)g * IN_DIM + DFEAT + lane * 4;
  atomicAdd(dst + 0, v.x);
  atomicAdd(dst + 1, v.y);
  atomicAdd(dst + 2, v.z);
  atomicAdd(dst + 3, v.w);
}

// ---------------------------------------------------------------------------
// Kernel 4: out[G,128] = relu(inp[G,384] @ W^T + b) via V_WMMA_F32_16X16X4_F32.
// One wave per 16x16 tile; K=384 in steps of 4.
//   A 16x4 layout: lane l -> m = l&15, VGPR{0,1} hold K = k0 + 2*(l>>4) + {0,1}
//   B 4x16 layout: lane l -> n = l&15, VGPR{0,1} hold K = k0 + 2*(l>>4) + {0,1}
//   C/D: VGPR v, lane l -> (m = 8*(l>>4) + v, n = l&15)
// ---------------------------------------------------------------------------
__global__ __launch_bounds__(32) void gemm_relu_wmma(const float* __restrict__ inp,
                                                     const float* __restrict__ W,
                                                     const float* __restrict__ bias,
                                                     float* __restrict__ out,
                                                     int out_dim) {
  const int tileN = blockIdx.x;          // out-feature tile
  const int tileM = blockIdx.y;          // graph tile
  const int lane = threadIdx.x;
  const int half = lane >> 4;            // 0 | 1
  const int l15 = lane & 15;
  const int mrow = tileM * 16 + l15;     // A row owned by this lane
  const int ncol = tileN * 16 + l15;     // B column owned by this lane

  const float* __restrict__ arow = inp + (size_t)mrow * IN_DIM;
  const float* __restrict__ brow = W + (size_t)ncol * IN_DIM; // B[k][n] = W[n*IN+k]

  v8f acc = {};
#pragma unroll 4
  for (int k0 = 0; k0 < IN_DIM; k0 += 4) {
    const int k = k0 + half * 2;
    v2f a, b;
    a.x = arow[k];
    a.y = arow[k + 1];
    b.x = brow[k];
    b.y = brow[k + 1];
    acc = __builtin_amdgcn_wmma_f32_16x16x4_f32(
        /*neg_a=*/false, a, /*neg_b=*/false, b,
        /*c_mod=*/(short)0, acc, /*reuse_a=*/false, /*reuse_b=*/false);
  }

  const float bv = bias[ncol];
#pragma unroll
  for (int v = 0; v < 8; ++v) {
    const int m = tileM * 16 + half * 8 + v;
    float r = acc[v] + bv;
    out[(size_t)m * out_dim + ncol] = r > 0.0f ? r : 0.0f;
  }
}

// ---------------------------------------------------------------------------
extern "C" void kernel_launch(void* const* d_in, const int* in_sizes, int n_in,
                              void* d_out, int out_size, void* d_ws, size_t ws_size,
                              hipStream_t stream) {
  const float* x     = (const float*)d_in[0];  // [N_NODES, 128]
  const int*   eidx  = (const int*)d_in[1];    // [2, N_EDGES] (row 0 = src)
  const float* eattr = (const float*)d_in[2];  // [N_EDGES, 128]
  const float* u     = (const float*)d_in[3];  // [G, 128]
  const int*   batch = (const int*)d_in[4];    // [N_NODES], sorted
  const float* W     = (const float*)d_in[5];  // [128, 384] row-major
  const float* b     = (const float*)d_in[6];  // [128]
  float* out = (float*)d_out;                  // [G, 128]
  float* inp = (float*)d_ws;                   // [G, 384] accumulator (1.5 MB)

  const int n_nodes = in_sizes[0] / DFEAT;
  const int n_edges = in_sizes[2] / DFEAT;
  const int G       = in_sizes[3] / DFEAT;     // 1024
  const int out_dim = out_size / G;            // 128

  const int total = G * IN_DIM;
  init_inp<<<(total + 255) / 256, 256, 0, stream>>>(u, inp, total);

  node_scatter<<<(n_nodes + 7) / 8, 256, 0, stream>>>(x, batch, inp, n_nodes);
  edge_scatter<<<(n_edges + 7) / 8, 256, 0, stream>>>(eattr, eidx, batch, inp, n_edges);

  dim3 grid(out_dim / 16, G / 16);
  gemm_relu_wmma<<<grid, 32, 0, stream>>>(inp, W, b, out, out_dim);
}